// GATModel_45191645888899
// MI455X (gfx1250) — compile-verified
//
#include <hip/hip_runtime.h>
#include <math.h>

// ---------------------------------------------------------------------------
// 2-layer GAT forward for MI455X (gfx1250, wave32).
// Dense linear layers use V_WMMA_F32_16X16X4_F32 (fp32 matrix core path) with
// compile-time K/Ncols so all inner-loop addressing folds into immediate
// offsets (no per-WMMA 64-bit address math), and a wave-uniform full-tile
// store fast path (no per-row EXEC juggling).
// Edge softmax/aggregation resolve in L2 (both 51 MB tables fit in 192 MB L2).
// ---------------------------------------------------------------------------

typedef __attribute__((ext_vector_type(2))) float v2f;
typedef __attribute__((ext_vector_type(8))) float v8f;

#define HEADS 4
#define HID   64
#define FEAT  256   // HEADS*HID
#define OUTC  8

__device__ __forceinline__ float elu1(float x)  { return x > 0.f ? x : expm1f(x); }
__device__ __forceinline__ float lrelu(float x) { return x > 0.f ? x : 0.2f * x; }

// Monotone bijection float -> uint so atomicMax(uint) == float max.
__device__ __forceinline__ unsigned fflip(float f) {
  unsigned u = __float_as_uint(f);
  return (u & 0x80000000u) ? ~u : (u | 0x80000000u);
}
__device__ __forceinline__ float funflip(unsigned u) {
  return __uint_as_float((u & 0x80000000u) ? (u & 0x7fffffffu) : ~u);
}

// edge_index is [2,E] row-major; self loops (i,i) appended logically at e>=Ein.
__device__ __forceinline__ void edge_sd(const int* __restrict__ ei, int e, int Ein,
                                        int* s, int* d) {
  if (e < Ein) { *s = ei[e]; *d = ei[Ein + e]; }
  else         { *s = *d = e - Ein; }
}

// ---------------------------------------------------------------------------
__global__ __launch_bounds__(256) void k_zero(float* __restrict__ p, long long n) {
  long long t = (long long)blockIdx.x * blockDim.x + threadIdx.x;
  if (t < n) p[t] = 0.f;
}

// C[M,NC] = A[M,K] @ B[K,NC], fp32 in/out, one wave per 16x16 tile.
// WMMA f32 16x16x4 lane layouts (ISA 7.12.2):
//   A[m][k]: lane = m + 16*(k>>1), reg = k&1
//   B[k][n]: lane = n + 16*(k>>1), reg = k&1   (B loaded "column-major" style)
//   C/D[m][n]: reg r holds m = 8*(lane>>4)+r, n = lane&15
// K, NC are template params so the unrolled loop uses immediate load offsets.
template <int K, int NC>
__global__ __launch_bounds__(256) void k_gemm_wmma(const float* __restrict__ A,
                                                   const float* __restrict__ B,
                                                   float* __restrict__ C, int M) {
  long long gtid = (long long)blockIdx.x * blockDim.x + threadIdx.x;
  int wave = (int)(gtid >> 5);
  int lane = threadIdx.x & 31;
  constexpr int tilesN = NC / 16;
  int tilesM = (M + 15) >> 4;
  if (wave >= tilesM * tilesN) return;      // wave-uniform: EXEC stays all-ones
  int tm = wave / tilesN, tn = wave % tilesN;

  int idx  = lane & 15;   // M index for A loads; N index for B loads & stores
  int half = lane >> 4;   // selects which K pair this lane carries

  int row = tm * 16 + idx; if (row >= M) row = M - 1;  // clamped load row
  const float* Ap = A + (size_t)row * K + 2 * half;            // + kb at use
  const float* Bp = B + (size_t)(2 * half) * NC + tn * 16 + idx; // + kb*NC at use

  v8f acc = {};
#pragma unroll 8
  for (int kb = 0; kb < K; kb += 4) {
    v2f a, b;
    a.x = Ap[kb];
    a.y = Ap[kb + 1];
    b.x = Bp[kb * NC];
    b.y = Bp[(kb + 1) * NC];
    acc = __builtin_amdgcn_wmma_f32_16x16x4_f32(false, a, false, b,
                                                (short)0, acc, false, false);
  }

  int mbase = tm * 16 + half * 8;
  int col   = tn * 16 + idx;
  float* Cp = C + (size_t)mbase * NC + col;
  if (tm * 16 + 16 <= M) {
    // Full tile (always taken for M % 16 == 0): unguarded stores, immediate
    // offsets r*NC*4 -> clause of 8 global_store_b32.
#pragma unroll
    for (int r = 0; r < 8; ++r) Cp[(size_t)r * NC] = acc[r];
  } else {
#pragma unroll
    for (int r = 0; r < 8; ++r)
      if (mbase + r < M) Cp[(size_t)r * NC] = acc[r];
  }
}

// alpha_src[n,h] = sum_c h[n,h,c]*a_src[h,c];  same for alpha_dst.
__global__ __launch_bounds__(256) void k_attn(const float* __restrict__ h,
                                              const float* __restrict__ a_s,
                                              const float* __restrict__ a_d,
                                              float* __restrict__ asrc,
                                              float* __restrict__ adst, int N) {
  int t = blockIdx.x * blockDim.x + threadIdx.x;
  if (t >= N * HEADS) return;
  int n = t >> 2, hd = t & 3;
  const float* hp = h + (size_t)n * FEAT + hd * HID;
  const float* sp = a_s + hd * HID;
  const float* dp = a_d + hd * HID;
  float ss = 0.f, sd = 0.f;
#pragma unroll 8
  for (int c = 0; c < HID; ++c) { float v = hp[c]; ss += v * sp[c]; sd += v * dp[c]; }
  asrc[t] = ss; adst[t] = sd;
}

__global__ __launch_bounds__(256) void k_edge_max(const int* __restrict__ ei,
                                                  const float* __restrict__ asrc,
                                                  const float* __restrict__ adst,
                                                  unsigned* __restrict__ emax,
                                                  int Ein, int Etot) {
  int t = blockIdx.x * blockDim.x + threadIdx.x;
  if (t >= Etot * HEADS) return;
  int e = t >> 2, hd = t & 3;
  int s, d; edge_sd(ei, e, Ein, &s, &d);
  float ev = lrelu(asrc[s * HEADS + hd] + adst[d * HEADS + hd]);
  atomicMax(&emax[d * HEADS + hd], fflip(ev));
}

__global__ __launch_bounds__(256) void k_edge_exp(const int* __restrict__ ei,
                                                  const float* __restrict__ asrc,
                                                  const float* __restrict__ adst,
                                                  const unsigned* __restrict__ emax,
                                                  float* __restrict__ alphaE,
                                                  float* __restrict__ denom,
                                                  int Ein, int Etot) {
  int t = blockIdx.x * blockDim.x + threadIdx.x;
  if (t >= Etot * HEADS) return;
  int e = t >> 2, hd = t & 3;
  int s, d; edge_sd(ei, e, Ein, &s, &d);
  float ev = lrelu(asrc[s * HEADS + hd] + adst[d * HEADS + hd]);
  float ee = __expf(ev - funflip(emax[d * HEADS + hd]));
  alphaE[t] = ee;
  atomicAdd(&denom[d * HEADS + hd], ee);
}

__global__ __launch_bounds__(256) void k_edge_norm(const int* __restrict__ ei,
                                                   float* __restrict__ alphaE,
                                                   const float* __restrict__ denom,
                                                   int Ein, int Etot) {
  int t = blockIdx.x * blockDim.x + threadIdx.x;
  if (t >= Etot * HEADS) return;
  int e = t >> 2, hd = t & 3;
  int s, d; edge_sd(ei, e, Ein, &s, &d);
  alphaE[t] = alphaE[t] / denom[d * HEADS + hd];
}

// out[dst,h,c] += alpha[e,h] * h[src,h,c]; thread = (edge, channel c in 0..63)
// so consecutive lanes hit consecutive addresses (coalesced; resolves in L2).
__global__ __launch_bounds__(256) void k_edge_aggr(const int* __restrict__ ei,
                                                   const float* __restrict__ h,
                                                   const float* __restrict__ alphaE,
                                                   float* __restrict__ out,
                                                   int Ein, int Etot) {
  long long t = (long long)blockIdx.x * blockDim.x + threadIdx.x;
  if (t >= (long long)Etot * HID) return;
  int e = (int)(t >> 6), c = (int)(t & 63);
  int s, d; edge_sd(ei, e, Ein, &s, &d);
  const float* hp = h + (size_t)s * FEAT + c;
  float* op = out + (size_t)d * FEAT + c;
#pragma unroll
  for (int hd = 0; hd < HEADS; ++hd)
    atomicAdd(op + hd * HID, hp[hd * HID] * alphaE[(size_t)e * HEADS + hd]);
}

__global__ __launch_bounds__(256) void k_bias_elu(float* __restrict__ buf,
                                                  const float* __restrict__ bias,
                                                  long long n) {
  long long t = (long long)blockIdx.x * blockDim.x + threadIdx.x;
  if (t >= n) return;
  int c = (int)(t & (FEAT - 1));
  buf[t] = elu1(buf[t] + bias[c]);
}

// z[n,c] = elu(mean_h(out2[n,h,c]) + b2[c])
__global__ __launch_bounds__(256) void k_mean_bias_elu(const float* __restrict__ o2,
                                                       const float* __restrict__ b2,
                                                       float* __restrict__ z, int N) {
  int t = blockIdx.x * blockDim.x + threadIdx.x;
  if (t >= N * HID) return;
  int n = t >> 6, c = t & 63;
  const float* p = o2 + (size_t)n * FEAT + c;
  float v = 0.25f * (p[0] + p[64] + p[128] + p[192]) + b2[c];
  z[t] = elu1(v);
}

// out[n,o] = sum_c z[n,c]*Wc[c,o] + bc[o]   (K=64, Nout=8 -> scalar is fine)
__global__ __launch_bounds__(256) void k_final(const float* __restrict__ z,
                                               const float* __restrict__ Wc,
                                               const float* __restrict__ bc,
                                               float* __restrict__ out, int N) {
  int t = blockIdx.x * blockDim.x + threadIdx.x;
  if (t >= N * OUTC) return;
  int n = t >> 3, o = t & 7;
  const float* zp = z + (size_t)n * HID;
  float acc = bc[o];
#pragma unroll 8
  for (int c = 0; c < HID; ++c) acc += zp[c] * Wc[c * OUTC + o];
  out[t] = acc;
}

// ---------------------------------------------------------------------------
extern "C" void kernel_launch(void* const* d_in, const int* in_sizes, int n_in,
                              void* d_out, int out_size, void* d_ws, size_t ws_size,
                              hipStream_t stream) {
  const float* x   = (const float*)d_in[0];
  const int*   ei  = (const int*)d_in[1];   // [2,E] (int32: JAX default x64-off)
  const float* W1  = (const float*)d_in[2];
  const float* as1 = (const float*)d_in[3];
  const float* ad1 = (const float*)d_in[4];
  const float* b1  = (const float*)d_in[5];
  const float* W2  = (const float*)d_in[6];
  const float* as2 = (const float*)d_in[7];
  const float* ad2 = (const float*)d_in[8];
  const float* b2  = (const float*)d_in[9];
  const float* Wc  = (const float*)d_in[10];
  const float* bc  = (const float*)d_in[11];
  float* out = (float*)d_out;

  const int N    = in_sizes[0] / 16;   // 50000
  const int Ein  = in_sizes[1] / 2;    // 800000
  const int Etot = Ein + N;            // + self loops

  // Workspace layout (~120 MB total)
  float*    bufA   = (float*)d_ws;                      // h1 / h2      [N*256]
  float*    bufB   = bufA + (size_t)N * FEAT;           // aggregate    [N*256]
  float*    asrc   = bufB + (size_t)N * FEAT;           // [N*4]
  float*    adst   = asrc + (size_t)N * HEADS;          // [N*4]
  float*    denom  = adst + (size_t)N * HEADS;          // [N*4]
  unsigned* emax   = (unsigned*)(denom + (size_t)N * HEADS); // [N*4]
  float*    alphaE = (float*)(emax + (size_t)N * HEADS);     // [Etot*4]

  const int TB = 256;
  auto NB = [](long long n) { return (unsigned)((n + 255) / 256); };

  const long long zspan = (long long)N * FEAT + (long long)N * HEADS * 4; // bufB..emax
  const long long nEH   = (long long)Etot * HEADS;
  const long long nEC   = (long long)Etot * HID;
  const long long waves = (long long)((N + 15) / 16) * (FEAT / 16);

  // ---------------- Layer 1 ----------------
  k_zero<<<NB(zspan), TB, 0, stream>>>(bufB, zspan);
  k_gemm_wmma<16, FEAT><<<NB(waves * 32), TB, 0, stream>>>(x, W1, bufA, N);
  k_attn<<<NB((long long)N * HEADS), TB, 0, stream>>>(bufA, as1, ad1, asrc, adst, N);
  k_edge_max <<<NB(nEH), TB, 0, stream>>>(ei, asrc, adst, emax, Ein, Etot);
  k_edge_exp <<<NB(nEH), TB, 0, stream>>>(ei, asrc, adst, emax, alphaE, denom, Ein, Etot);
  k_edge_norm<<<NB(nEH), TB, 0, stream>>>(ei, alphaE, denom, Ein, Etot);
  k_edge_aggr<<<NB(nEC), TB, 0, stream>>>(ei, bufA, alphaE, bufB, Ein, Etot);
  k_bias_elu <<<NB((long long)N * FEAT), TB, 0, stream>>>(bufB, b1, (long long)N * FEAT);

  // ---------------- Layer 2 ----------------
  k_gemm_wmma<FEAT, FEAT><<<NB(waves * 32), TB, 0, stream>>>(bufB, W2, bufA, N);
  k_zero<<<NB(zspan), TB, 0, stream>>>(bufB, zspan);   // after gemm consumed bufB
  k_attn<<<NB((long long)N * HEADS), TB, 0, stream>>>(bufA, as2, ad2, asrc, adst, N);
  k_edge_max <<<NB(nEH), TB, 0, stream>>>(ei, asrc, adst, emax, Ein, Etot);
  k_edge_exp <<<NB(nEH), TB, 0, stream>>>(ei, asrc, adst, emax, alphaE, denom, Ein, Etot);
  k_edge_norm<<<NB(nEH), TB, 0, stream>>>(ei, alphaE, denom, Ein, Etot);
  k_edge_aggr<<<NB(nEC), TB, 0, stream>>>(ei, bufA, alphaE, bufB, Ein, Etot);

  // mean over heads + b2 + ELU -> z (reuse bufA), then final projection
  k_mean_bias_elu<<<NB((long long)N * HID), TB, 0, stream>>>(bufB, b2, bufA, N);
  k_final<<<NB((long long)N * OUTC), TB, 0, stream>>>(bufA, Wc, bc, out, N);
}